// Sim_16492674417472
// MI455X (gfx1250) — compile-verified
//
#include <hip/hip_runtime.h>
#include <hip/hip_bf16.h>

typedef __attribute__((ext_vector_type(2))) float v2f;
typedef __attribute__((ext_vector_type(4))) float v4f;
typedef __attribute__((ext_vector_type(8))) float v8f;

#define BN_EPS 1e-3f

// ---------------------------------------------------------------------------
// Kernel 1: 1x1-conv GEMM + folded-BN + SiLU, fp32 WMMA (V_WMMA_F32_16X16X4_F32)
//   Y[bt, o, hw] = silu( inv[o] * (sum_c W[o,c] * X[bt, c, hw]) + bias[o] )
// Block = 128 threads (4 wave32). Block tile = 64(O) x 64(N). Each wave owns a
// 16(O)-row stripe and all four 16-wide N subtiles (4 x v8f accumulators).
// Fragment layouts per CDNA5 ISA 7.12.2:
//   A (16x4): lane l -> M = l%16, K = 2*(l/16) + vgpr
//   B (4x16): vgpr r, lane l -> K = r + 2*(l/16), N = l%16
//   C/D     : vgpr r, lane l -> M = r + 8*(l/16), N = l%16
// X tile is stored in LDS interleaved by K-pair:
//   Xs[((k>>1)*64 + n)*2 + (k&1)]
// so a B fragment (two adjacent-K values, same N) is ONE 8B-aligned
// ds_load_b64 directly into the even-aligned VGPR pair the WMMA consumes.
// ---------------------------------------------------------------------------
__global__ __launch_bounds__(128)
void conv_bn_silu_kernel(const float* __restrict__ X,   // [B*T, C, HW]
                         const float* __restrict__ Wm,  // [O=C, C]
                         const float* __restrict__ gg,
                         const float* __restrict__ bb,
                         const float* __restrict__ mm,
                         const float* __restrict__ vv,
                         float* __restrict__ Y,         // [B*T, O, HW]
                         int C, int HW)
{
    __shared__ float Ws[64 * 16];   // [oo][kk], row-major
    __shared__ float Xs[16 * 64];   // [kpair][n][kin] interleaved

    const int tid  = threadIdx.x;
    const int wave = tid >> 5;
    const int lane = tid & 31;
    const int lm   = lane & 15;     // low-half lane id
    const int lh   = lane >> 4;     // half-wave select

    const int bt = blockIdx.z;
    const int o0 = blockIdx.y * 64;
    const int n0 = blockIdx.x * 64;

    const float* Xb = X + (size_t)bt * C * HW;
    float*       Yb = Y + (size_t)bt * C * HW;   // O == C

    // X staging coords for this thread: one (kpair, n4) unit = 2 rows x 4 cols.
    const int skp = tid >> 4;       // kpair 0..7
    const int sn4 = tid & 15;       // n4    0..15

    const v8f zero8 = {0.f,0.f,0.f,0.f,0.f,0.f,0.f,0.f};
    v8f acc[4];
#pragma unroll
    for (int i = 0; i < 4; ++i) acc[i] = zero8;

    for (int k0 = 0; k0 < C; k0 += 16) {
        __syncthreads();
        // Stage W tile 64x16: 256 float4s, 2 per thread.
#pragma unroll
        for (int i = 0; i < 2; ++i) {
            int idx = tid + i * 128;
            int oo  = idx >> 2, kk4 = idx & 3;
            v4f wv = *(const v4f*)&Wm[(size_t)(o0 + oo) * C + k0 + kk4 * 4];
            *(v4f*)&Ws[oo * 16 + kk4 * 4] = wv;
        }
        // Stage X tile 16x64 interleaved by K-pair: 1 unit per thread.
        {
            v4f x0 = *(const v4f*)&Xb[(size_t)(k0 + 2 * skp + 0) * HW + n0 + sn4 * 4];
            v4f x1 = *(const v4f*)&Xb[(size_t)(k0 + 2 * skp + 1) * HW + n0 + sn4 * 4];
#pragma unroll
            for (int j = 0; j < 4; ++j) {
                v2f p; p.x = x0[j]; p.y = x1[j];
                *(v2f*)&Xs[(skp * 64 + sn4 * 4 + j) * 2] = p;
            }
        }
        __syncthreads();

        // Hint the next X K-tile into cache (global_prefetch_b8).
        if (k0 + 16 < C)
            __builtin_prefetch(&Xb[(size_t)(k0 + 16 + (tid >> 4)) * HW + n0 + (tid & 15) * 4], 0, 3);

        // 16 WMMAs per K-step: 4 K-chunks of 4 x 4 N-subtiles.
#pragma unroll
        for (int ks = 0; ks < 4; ++ks) {
            v2f a = *(const v2f*)&Ws[(wave * 16 + lm) * 16 + ks * 4 + 2 * lh];
            const int kpair = ks * 2 + lh;      // (ks*4 + 2*lh) >> 1
#pragma unroll
            for (int n16 = 0; n16 < 4; ++n16) {
                v2f bfrag = *(const v2f*)&Xs[(kpair * 64 + n16 * 16 + lm) * 2];
                acc[n16] = __builtin_amdgcn_wmma_f32_16x16x4_f32(
                    false, a, false, bfrag, (short)0, acc[n16], false, false);
            }
        }
    }

    // Epilogue: folded BN + SiLU, per C/D layout.
#pragma unroll
    for (int r = 0; r < 8; ++r) {
        int o = o0 + wave * 16 + r + 8 * lh;
        float inv = gg[o] * rsqrtf(vv[o] + BN_EPS);
        float bia = bb[o] - mm[o] * inv;
#pragma unroll
        for (int n16 = 0; n16 < 4; ++n16) {
            float y = acc[n16][r] * inv + bia;
            float s = y / (1.0f + __expf(-y));     // y * sigmoid(y)
            Yb[(size_t)o * HW + n0 + n16 * 16 + lm] = s;
        }
    }
}

// ---------------------------------------------------------------------------
// Kernel 2: local correlation. One block per (b, c) plane. Threads 0..242 own
// a (t, shift) pair; frame-3 tile (with halo) and frames 0..2 staged in LDS.
//   sims[b,c,t, dy*9+dx] = sum_{h,w} f1[h+(dy-4)s, w+(dx-4)s] * f2[t,h,w]
// ---------------------------------------------------------------------------
#define CORR_TH 8
__global__ __launch_bounds__(256)
void corr_kernel(const float* __restrict__ PF,   // [B, T=4, C, H, W]
                 float* __restrict__ sims,       // [B, C, 3, 81]
                 int C, int H, int W, int s)
{
    __shared__ float f1s[40 * 128];              // worst case: (8+32) x (96+32)
    __shared__ float f2s[3 * CORR_TH * 96];      // worst case: 3 x 8 x 96

    const int tid = threadIdx.x;
    const int c   = blockIdx.x;
    const int b   = blockIdx.y;
    const int HW  = H * W;
    const int p   = 4 * s;
    const int f1W = W + 2 * p;
    const int f1H = CORR_TH + 2 * p;

    const float* f1     = PF + (((size_t)b * 4 + 3) * C + c) * HW;   // frame T-1
    const float* f2base = PF + ((size_t)b * 4 * C + c) * HW;         // + t*C*HW

    const int t  = tid / 81;
    const int k  = tid % 81;
    const int dy = k / 9, dx = k % 9;
    float acc = 0.f;

    for (int h0 = 0; h0 < H; h0 += CORR_TH) {
        __syncthreads();
        // frame-3 tile with halo, zero-padded
        for (int idx = tid; idx < f1H * f1W; idx += 256) {
            int hh = idx / f1W, ww = idx % f1W;
            int gh = h0 + hh - p, gw = ww - p;
            float val = 0.f;
            if (gh >= 0 && gh < H && gw >= 0 && gw < W) val = f1[gh * W + gw];
            f1s[idx] = val;
        }
        // frames 0..2, CORR_TH rows
        for (int idx = tid; idx < 3 * CORR_TH * W; idx += 256) {
            int tt  = idx / (CORR_TH * W);
            int rem = idx % (CORR_TH * W);
            f2s[idx] = f2base[(size_t)tt * C * HW + (h0 + rem / W) * W + rem % W];
        }
        __syncthreads();
        if (tid < 243) {
            for (int r = 0; r < CORR_TH; ++r) {
                const float* l1 = &f1s[(r + dy * s) * f1W + dx * s];
                const float* l2 = &f2s[(t * CORR_TH + r) * W];
                for (int w = 0; w < W; ++w) acc = fmaf(l1[w], l2[w], acc);
            }
        }
    }
    if (tid < 243)
        sims[((size_t)(b * C + c) * 3 + t) * 81 + k] = acc;
}

// ---------------------------------------------------------------------------
// Kernel 3: softmax over 81 shifts, scale by future/-past, dot with MLP.
//   out[b,c] = sum_t scale[b,t] * sum_k softmax_k(sim[b,c,t,:]) * mlp[t*81+k]
// ---------------------------------------------------------------------------
__global__ __launch_bounds__(256)
void finalize_kernel(const float* __restrict__ sims,  // [B, C, 3, 81]
                     const float* __restrict__ mlp,   // [243]
                     const int*  __restrict__ past,   // [B, 4]
                     const int*  __restrict__ fut,    // [B, 1]
                     float* __restrict__ out,         // [B, C]
                     int C)
{
    int gid = blockIdx.x * 256 + threadIdx.x;
    if (gid >= 4 * C) return;
    int b = gid / C, c = gid % C;
    const float* sp = sims + (size_t)(b * C + c) * 243;
    float accum = 0.f;
    for (int t = 0; t < 3; ++t) {
        float scale = (float)fut[b] / (float)(-past[b * 4 + t]);
        const float* sv = sp + t * 81;
        float mx = sv[0];
        for (int kk = 1; kk < 81; ++kk) mx = fmaxf(mx, sv[kk]);
        float sum = 0.f, dot = 0.f;
        for (int kk = 0; kk < 81; ++kk) {
            float e = __expf(sv[kk] - mx);
            sum += e;
            dot += e * mlp[t * 81 + kk];
        }
        accum += scale * dot / sum;
    }
    out[(size_t)b * C + c] = accum;
}

// ---------------------------------------------------------------------------
// Host launch
// Input order (setup_inputs dict order):
//   [l*6+0]=feat_l [l*6+1]=w_l [l*6+2]=g_l [l*6+3]=b_l [l*6+4]=m_l [l*6+5]=v_l
//   [18]=mlp_w  [19]=past_clip_ids (int)  [20]=future_clip_ids (int)
// ---------------------------------------------------------------------------
extern "C" void kernel_launch(void* const* d_in, const int* in_sizes, int n_in,
                              void* d_out, int out_size, void* d_ws, size_t ws_size,
                              hipStream_t stream) {
    (void)in_sizes; (void)n_in; (void)out_size; (void)ws_size;
    const int B = 4, T = 4;
    const int Cs[3] = {256, 512, 1024};
    const int Hs[3] = {64, 32, 16};
    const int Wd[3] = {96, 48, 24};
    const int Ss[3] = {4, 2, 1};

    float* ws = (float*)d_ws;
    size_t pfOff[3], simOff[3], cur = 0;
    for (int l = 0; l < 3; ++l) { pfOff[l] = cur; cur += (size_t)B * T * Cs[l] * Hs[l] * Wd[l]; }
    for (int l = 0; l < 3; ++l) { simOff[l] = cur; cur += (size_t)B * Cs[l] * 243; }

    const float* mlp  = (const float*)d_in[18];
    const int*   past = (const int*)d_in[19];
    const int*   fut  = (const int*)d_in[20];

    size_t outOff = 0;
    for (int l = 0; l < 3; ++l) {
        const int C = Cs[l], H = Hs[l], W = Wd[l], HW = H * W;
        const float* feat = (const float*)d_in[l * 6 + 0];
        const float* Wm   = (const float*)d_in[l * 6 + 1];
        const float* gg   = (const float*)d_in[l * 6 + 2];
        const float* bb   = (const float*)d_in[l * 6 + 3];
        const float* mm   = (const float*)d_in[l * 6 + 4];
        const float* vv   = (const float*)d_in[l * 6 + 5];

        dim3 gConv(HW / 64, C / 64, B * T);
        conv_bn_silu_kernel<<<gConv, 128, 0, stream>>>(
            feat, Wm, gg, bb, mm, vv, ws + pfOff[l], C, HW);

        dim3 gCorr(C, B);
        corr_kernel<<<gCorr, 256, 0, stream>>>(
            ws + pfOff[l], ws + simOff[l], C, H, W, Ss[l]);

        int nfin = (B * C + 255) / 256;
        finalize_kernel<<<nfin, 256, 0, stream>>>(
            ws + simOff[l], mlp, past, fut, (float*)d_out + outOff, C);

        outOff += (size_t)B * C;
    }
}